// kuramoto_angular_closed_loop_19636590477624
// MI455X (gfx1250) — compile-verified
//
#include <hip/hip_runtime.h>
#include <stdint.h>

#define N_NODES 4096
#define COUPLING 3.0f
#define POWER_ITERS 10
#define KSPLIT 8          // K-dimension split for occupancy
#define KC 32             // floats per staged k-chunk
#define TILE_M 16         // rows per wave (one WMMA tile)

typedef float v2f __attribute__((ext_vector_type(2)));
typedef float v8f __attribute__((ext_vector_type(8)));
typedef int   v4i __attribute__((ext_vector_type(4)));

#if defined(__AMDGCN__) && __has_builtin(__builtin_amdgcn_global_load_async_to_lds_b128) && __has_builtin(__builtin_amdgcn_s_wait_asynccnt)
#define USE_ASYNC_LDS 1
#else
#define USE_ASYNC_LDS 0
#endif

#if USE_ASYNC_LDS
#define GLOBAL_AS __attribute__((address_space(1)))
#define LDS_AS __attribute__((address_space(3)))
// Flat addresses in the LDS aperture map to LDS by truncation to addr[31:0]
// (ISA 10.2), so integer casts give us the AS-qualified pointers the builtin
// wants without an addrspacecast. Builtin signature (from the compiler
// diagnostic): (v4i __device__* gsrc, v4i __shared__* ldst, imm off, imm cpol).
__device__ __forceinline__ void async_ld16(const float* g, const float* l) {
    __builtin_amdgcn_global_load_async_to_lds_b128(
        (GLOBAL_AS v4i*)(uintptr_t)g,
        (LDS_AS v4i*)(uint32_t)(uintptr_t)l, 0, 0);
}
#endif

// ---------------------------------------------------------------------------
// Kernel 1: per-node prep. Builds the 16-wide RHS block V:
//   col0=w, col1=cos(x1), col2=sin(x1), col3=cos*x2, col4=sin*x2, col5=x2
// plus dH2, sw, and initializes the power-iteration buffers.
// ---------------------------------------------------------------------------
__global__ __launch_bounds__(256) void prep_kernel(
    const float* __restrict__ x, const float* __restrict__ Km,
    const float* __restrict__ b, const float* __restrict__ G,
    float* __restrict__ V, float* __restrict__ PV, float* __restrict__ PZ,
    float* __restrict__ DH, float* __restrict__ SW, int n)
{
    int i = blockIdx.x * blockDim.x + threadIdx.x;
    if (i >= n) return;

    float x1  = x[i];
    float x2  = x[n + i];
    float xi0 = x[2 * n + 2 * i + 0];
    float xi1 = x[2 * n + 2 * i + 1];
    float K00 = Km[4 * i + 0], K01 = Km[4 * i + 1];
    float K10 = Km[4 * i + 2], K11 = Km[4 * i + 3];
    float b0 = b[2 * i], b1 = b[2 * i + 1];
    float g0 = G[2 * i], g1 = G[2 * i + 1];

    // M = tanh(K @ xi + b);  dH2 = K^T @ M
    float m0 = tanhf(fmaf(K00, xi0, fmaf(K01, xi1, b0)));
    float m1 = tanhf(fmaf(K10, xi0, fmaf(K11, xi1, b1)));
    float dh0 = fmaf(K00, m0, K10 * m1);
    float dh1 = fmaf(K01, m0, K11 * m1);
    float w   = fmaf(g0, dh0, g1 * dh1);

    float s, c;
    sincosf(x1, &s, &c);

    float* vr = V + (size_t)i * 16;
    vr[0] = w;  vr[1] = c;  vr[2] = s;
    vr[3] = c * x2;  vr[4] = s * x2;  vr[5] = x2;
#pragma unroll
    for (int j = 6; j < 16; ++j) vr[j] = 0.f;

    float* pv = PV + (size_t)i * 16;
    float* pz = PZ + (size_t)i * 16;
    pv[0] = 1.f;
#pragma unroll
    for (int j = 1; j < 16; ++j) pv[j] = 0.f;
#pragma unroll
    for (int j = 0; j < 16; ++j) pz[j] = 0.f;

    DH[2 * i + 0] = dh0;
    DH[2 * i + 1] = dh1;
    float inv_n2 = 1.0f / ((float)n * (float)n);
    SW[i] = fmaf(g0, g0, g1 * g1) * inv_n2;
}

// ---------------------------------------------------------------------------
// Kernel 2: K-split GEMM partials  P[seg] = A[:, seg] @ V[seg, :]
// via V_WMMA_F32_16X16X4_F32.
//  - block = 4 waves, each wave owns one 16-row tile (wave-private LDS, no
//    barriers); grid = (n/64, KSPLIT) -> 2048 waves for latency hiding.
//  - A chunks (16 x KC floats) staged through LDS, double buffered, with
//    coalesced b128 loads; CDNA5 async-to-LDS path when available
//    (ASYNCcnt + s_wait_asynccnt), synchronous ds_store fallback otherwise
//    (DS ops are in-order per wave, so no explicit waits needed).
//  - A-frag layout (ISA 7.12.2, 32-bit A 16x4): lanes 0-15 K={0,1},
//    lanes 16-31 K={2,3} -> per-lane ds_load_b64 of 2 consecutive floats.
// ---------------------------------------------------------------------------
__global__ __launch_bounds__(128) void wmma_gemm16_split_kernel(
    const float* __restrict__ A, const float* __restrict__ V,
    float* __restrict__ P, int n)
{
    __shared__ float lds[4 * 2 * TILE_M * KC];   // 4 waves x 2 bufs x 16x32

    const int lane = threadIdx.x & 31;
    const int wave = threadIdx.x >> 5;
    const int tile = blockIdx.x * 4 + wave;
    const int row0 = tile * TILE_M;
    const int seg  = blockIdx.y;
    const int segL = n / KSPLIT;
    const int k0   = seg * segL;
    const int nchunks = segL / KC;

    const int mr = lane & 15;   // A row within tile / B,C column
    const int hi = lane >> 4;   // half-wave select

    float* wlds = lds + wave * (2 * TILE_M * KC);

    // Staging map: lane covers flat floats [lane*16, lane*16+16) of the
    // row-major 16xKC chunk -> row = lane/2, cols = (lane&1)*16 .. +16.
    const int sr = lane >> 1;
    const int sc = (lane & 1) * 16;
    const float* gstage = A + (size_t)(row0 + sr) * (size_t)n + k0 + sc;
    float* lstage = wlds + lane * 16;

#define STAGE_CHUNK(cc, bb)                                                   \
    do {                                                                      \
        const float* gs = gstage + (size_t)(cc) * KC;                         \
        float* ls = lstage + (bb) * (TILE_M * KC);                            \
        STAGE_ONE(gs, ls, 0);                                                 \
        STAGE_ONE(gs, ls, 1);                                                 \
        STAGE_ONE(gs, ls, 2);                                                 \
        STAGE_ONE(gs, ls, 3);                                                 \
    } while (0)

#if USE_ASYNC_LDS
#define STAGE_ONE(gs, ls, q) async_ld16((gs) + (q) * 4, (ls) + (q) * 4)
#else
#define STAGE_ONE(gs, ls, q) \
    *(float4*)((ls) + (q) * 4) = *(const float4*)((gs) + (q) * 4)
#endif

    // prologue: stage chunk 0 into buffer 0
    STAGE_CHUNK(0, 0);

    v8f acc0 = {0.f, 0.f, 0.f, 0.f, 0.f, 0.f, 0.f, 0.f};
    v8f acc1 = {0.f, 0.f, 0.f, 0.f, 0.f, 0.f, 0.f, 0.f};

#pragma unroll 2
    for (int c = 0; c < nchunks; ++c) {
        const int buf = c & 1;
        if (c + 1 < nchunks) {
            STAGE_CHUNK(c + 1, buf ^ 1);
#if USE_ASYNC_LDS
            // 4 async b128 per chunk, in-order completion: <=4 outstanding
            // means the current chunk has fully landed in LDS.
            __builtin_amdgcn_s_wait_asynccnt(4);
#endif
        } else {
#if USE_ASYNC_LDS
            __builtin_amdgcn_s_wait_asynccnt(0);
#endif
        }

        const float* t  = wlds + buf * (TILE_M * KC) + mr * KC + 2 * hi;
        const float* Vb = V + (size_t)(k0 + c * KC + 2 * hi) * 16 + mr;

#pragma unroll
        for (int kk = 0; kk < KC; kk += 8) {
            v2f a0 = *(const v2f*)(t + kk);
            v2f b0;
            b0.x = Vb[(size_t)(kk + 0) * 16];
            b0.y = Vb[(size_t)(kk + 1) * 16];
            acc0 = __builtin_amdgcn_wmma_f32_16x16x4_f32(
                false, a0, false, b0, (short)0, acc0, false, false);

            v2f a1 = *(const v2f*)(t + kk + 4);
            v2f b1;
            b1.x = Vb[(size_t)(kk + 4) * 16];
            b1.y = Vb[(size_t)(kk + 5) * 16];
            acc1 = __builtin_amdgcn_wmma_f32_16x16x4_f32(
                false, a1, false, b1, (short)0, acc1, false, false);
        }
    }

    v8f acc = acc0 + acc1;
    // C/D layout: VGPR j -> (M=j, N=lane) lanes 0-15; (M=j+8, N=lane-16) 16-31
    float* pr = P + (size_t)seg * (size_t)n * 16
                  + (size_t)(row0 + hi * 8) * 16 + mr;
#pragma unroll
    for (int j = 0; j < 8; ++j) pr[(size_t)j * 16] = acc[j];
#undef STAGE_CHUNK
#undef STAGE_ONE
}

// Deterministic fixed-order sum of the KSPLIT partial GEMM results.
__global__ __launch_bounds__(256) void reduce_partials_kernel(
    const float* __restrict__ P, float* __restrict__ Y, int count)
{
    int i = blockIdx.x * blockDim.x + threadIdx.x;
    if (i >= count) return;
    float s = 0.f;
#pragma unroll
    for (int q = 0; q < KSPLIT; ++q) s += P[(size_t)q * count + i];
    Y[i] = s;
}

// ---------------------------------------------------------------------------
// Power-iteration helpers (all deterministic, fixed-shape)
// ---------------------------------------------------------------------------
__global__ __launch_bounds__(256) void pscale_kernel(
    const float* __restrict__ PY, const float* __restrict__ SW,
    float* __restrict__ PZ, int n)
{
    int i = blockIdx.x * blockDim.x + threadIdx.x;
    if (i < n) PZ[(size_t)i * 16] = SW[i] * PY[(size_t)i * 16];
}

__global__ __launch_bounds__(256) void rnorm1_kernel(
    const float* __restrict__ PY, float* __restrict__ part, int n)
{
    __shared__ float sm[256];
    int i = blockIdx.x * 256 + threadIdx.x;
    float v = (i < n) ? PY[(size_t)i * 16] : 0.f;
    sm[threadIdx.x] = v * v;
    __syncthreads();
#pragma unroll
    for (int s = 128; s > 0; s >>= 1) {
        if (threadIdx.x < s) sm[threadIdx.x] += sm[threadIdx.x + s];
        __syncthreads();
    }
    if (threadIdx.x == 0) part[blockIdx.x] = sm[0];
}

__global__ void rnorm2_kernel(const float* __restrict__ part,
                              float* __restrict__ SC, int nb)
{
    if (threadIdx.x == 0) {
        float s = 0.f;
        for (int j = 0; j < nb; ++j) s += part[j];
        float lam = sqrtf(s);
        SC[0] = s;
        SC[1] = (s > 0.f) ? rsqrtf(s) : 0.f;
        SC[2] = lam;
        SC[3] = 0.85f * lam;   // gamma (last iteration wins)
    }
}

__global__ __launch_bounds__(256) void pnorm_kernel(
    const float* __restrict__ PY, const float* __restrict__ SC,
    float* __restrict__ PV, int n)
{
    int i = blockIdx.x * blockDim.x + threadIdx.x;
    if (i < n) PV[(size_t)i * 16] = PY[(size_t)i * 16] * SC[1];
}

// ---------------------------------------------------------------------------
// Kernel 3: finalize. Recombines the 6 mat-vec results into the output.
// ---------------------------------------------------------------------------
__global__ __launch_bounds__(256) void finalize_kernel(
    const float* __restrict__ V, const float* __restrict__ Y,
    const float* __restrict__ DH, const float* __restrict__ G,
    const float* __restrict__ SC, float* __restrict__ out, int n)
{
    int j = blockIdx.x * blockDim.x + threadIdx.x;
    if (j >= n) return;
    const float* vr = V + (size_t)j * 16;
    const float* yr = Y + (size_t)j * 16;
    float c = vr[1], s = vr[2], x2 = vr[5];
    float u = -yr[0];
    // s_int[j] = c*x2*(A c) + s*x2*(A s) - c*(A cx2) - s*(A sx2)
    float s_int = c * x2 * yr[1] + s * x2 * yr[2] - c * yr[3] - s * yr[4];
    float inv_n = 1.f / (float)n;
    float dxdt = COUPLING * inv_n * u * s_int;

    float gamma = SC[3];
    float dh0 = DH[2 * j], dh1 = DH[2 * j + 1];
    float g0 = G[2 * j], g1 = G[2 * j + 1];
    float Ax2 = yr[5];
    float dxi0 = -dh1 - gamma * dh0 + g0 * inv_n * Ax2;
    float dxi1 =  dh0 - gamma * dh1 + g1 * inv_n * Ax2;

    out[j] = x2;
    out[n + j] = dxdt;
    out[2 * n + 2 * j + 0] = dxi0;
    out[2 * n + 2 * j + 1] = dxi1;
}

// ---------------------------------------------------------------------------
extern "C" void kernel_launch(void* const* d_in, const int* in_sizes, int n_in,
                              void* d_out, int out_size, void* d_ws, size_t ws_size,
                              hipStream_t stream)
{
    (void)in_sizes; (void)n_in; (void)out_size; (void)ws_size;
    const int n = N_NODES;
    const float* x   = (const float*)d_in[1];
    const float* adj = (const float*)d_in[2];
    const float* Km  = (const float*)d_in[3];
    const float* b   = (const float*)d_in[4];
    const float* G   = (const float*)d_in[5];
    float* out = (float*)d_out;

    float* ws = (float*)d_ws;
    float* V    = ws;                     // n*16
    float* Y    = V  + (size_t)n * 16;    // n*16
    float* PV   = Y  + (size_t)n * 16;    // n*16
    float* PY   = PV + (size_t)n * 16;    // n*16
    float* PZ   = PY + (size_t)n * 16;    // n*16
    float* DH   = PZ + (size_t)n * 16;    // 2n
    float* SW   = DH + (size_t)2 * n;     // n
    float* PART = SW + n;                 // n/256 partials
    float* SC   = PART + (n / 256);       // 4 scalars
    float* P    = SC + 4;                 // KSPLIT * n * 16 GEMM partials

    dim3 b256(256), g16(n / 256);
    dim3 gGemm(n / 64, KSPLIT), bGemm(128);
    const int cnt = n * 16;
    dim3 gRed(cnt / 256);

    prep_kernel<<<g16, b256, 0, stream>>>(x, Km, b, G, V, PV, PZ, DH, SW, n);

    // 6 fused mat-vecs: Y = A @ V  (one pass over the 67MB adjacency)
    wmma_gemm16_split_kernel<<<gGemm, bGemm, 0, stream>>>(adj, V, P, n);
    reduce_partials_kernel<<<gRed, b256, 0, stream>>>(P, Y, cnt);

    // gamma = 0.85 * lambda_max(A^T diag(sw) A) by power iteration
    for (int it = 0; it < POWER_ITERS; ++it) {
        wmma_gemm16_split_kernel<<<gGemm, bGemm, 0, stream>>>(adj, PV, P, n);
        reduce_partials_kernel<<<gRed, b256, 0, stream>>>(P, PY, cnt);   // y = A v
        pscale_kernel<<<g16, b256, 0, stream>>>(PY, SW, PZ, n);          // z = sw.*y
        wmma_gemm16_split_kernel<<<gGemm, bGemm, 0, stream>>>(adj, PZ, P, n);
        reduce_partials_kernel<<<gRed, b256, 0, stream>>>(P, PY, cnt);   // y = A z = M v
        rnorm1_kernel<<<dim3(n / 256), b256, 0, stream>>>(PY, PART, n);
        rnorm2_kernel<<<1, 32, 0, stream>>>(PART, SC, n / 256);
        pnorm_kernel<<<g16, b256, 0, stream>>>(PY, SC, PV, n);           // v = y/|y|
    }

    finalize_kernel<<<g16, b256, 0, stream>>>(V, Y, DH, G, SC, out, n);
}